// LocalStridedBlockSparseAttnInferenceBT_81011673137832
// MI455X (gfx1250) — compile-verified
//
#include <hip/hip_runtime.h>
#include <math.h>

typedef float v2f __attribute__((ext_vector_type(2)));
typedef float v4f __attribute__((ext_vector_type(4)));
typedef float v8f __attribute__((ext_vector_type(8)));
typedef int   v4i __attribute__((ext_vector_type(4)));

#define B_    16
#define H_    32
#define KV_   8
#define D_    128
#define BS_   16
#define NBLK_ 256
#define GQA_  4          // H_/KV_
#define PATH_ (NBLK_*NBLK_)
#define NW_   8          // waves per workgroup

#define AS1 __attribute__((address_space(1)))
#define AS3 __attribute__((address_space(3)))

#if defined(__HIP_DEVICE_COMPILE__) && __has_builtin(__builtin_amdgcn_global_load_async_to_lds_b128)
#define ASYNC_V 1
#else
#define ASYNC_V 0
#endif

// One workgroup per (batch, kv_head). 8 waves, flash-decoding split over
// active sparse blocks, cross-wave softmax merge at the end.
__global__ __launch_bounds__(32 * NW_) void sparse_decode_attn(
    const float* __restrict__ q,
    const float* __restrict__ kc,
    const float* __restrict__ vc,
    const int*   __restrict__ bt,
    const int*   __restrict__ ctx,
    const unsigned char* __restrict__ pat,
    float* __restrict__ out)
{
    __shared__ float qlds[16 * 128];          // 16 rows (4 heads + zero pad) x 128
    __shared__ float vlds[NW_][128 * 16];     // per-wave V tile, [d][t]
    __shared__ float plds[NW_][16 * 16];      // per-wave P tile, [head-row][t]
    __shared__ float outbuf[GQA_ * 128];
    __shared__ float mx[NW_][GQA_];
    __shared__ float ls[NW_][GQA_];

    const int tid  = threadIdx.x;
    const int wave = tid >> 5;
    const int lane = tid & 31;
    const int b    = blockIdx.x / KV_;
    const int kvh  = blockIdx.x % KV_;
    const int h0   = kvh * GQA_;

    const float sm_scale = 0.08838834764831845f;   // 1/sqrt(128)

    // ---- init LDS: Q rows (scaled), zero pad rows, zero output buf + P tiles
    for (int i = tid; i < 16 * 128; i += 32 * NW_) {
        int row = i >> 7, d = i & 127;
        qlds[i] = (row < GQA_) ? q[((size_t)b * H_ + h0 + row) * D_ + d] * sm_scale : 0.0f;
    }
    for (int i = tid; i < GQA_ * 128; i += 32 * NW_) outbuf[i] = 0.0f;
    if (lane < 16) {
        for (int i = lane; i < 256; i += 16) plds[wave][i] = 0.0f;
    }
    __syncthreads();

    const int past_len = ctx[b] - 1;
    const int qpb      = past_len >> 4;            // query's block row in pattern

    const int tok  = lane & 15;                    // token / column index
    const int xoff = (lane < 16) ? 0 : 2;          // K-dim pair selector (WMMA layout)

    float m[GQA_], l[GQA_];
    v8f acc[8];
    for (int j = 0; j < GQA_; ++j) { m[j] = -INFINITY; l[j] = 0.0f; }
    for (int n = 0; n < 8; ++n) acc[n] = (v8f){0,0,0,0,0,0,0,0};

    const unsigned char* prow = pat + (size_t)h0 * PATH_ + (size_t)qpb * NBLK_;

    int aidx = 0;
    for (int kb = 0; kb <= qpb; ++kb) {
        unsigned char p0 = prow[kb];
        unsigned char p1 = prow[PATH_ + kb];
        unsigned char p2 = prow[2 * PATH_ + kb];
        unsigned char p3 = prow[3 * PATH_ + kb];
        if (!(p0 | p1 | p2 | p3)) continue;              // inactive for all 4 heads
        bool mine = ((aidx & (NW_ - 1)) == wave); ++aidx; // round-robin active blocks
        if (!mine) continue;

        const int phys = bt[b * NBLK_ + kb];
        const float* kbase = kc + (size_t)(phys * KV_ + kvh) * 2048;  // [d/4][t][4]
        const float* vbase = vc + (size_t)(phys * KV_ + kvh) * 2048;  // [d][t]

        // Pull the K tile toward the caches ahead of the streaming loop.
        __builtin_prefetch(kbase + lane * 64, 0, 3);

#if ASYNC_V
        // ---------- kick off async V tile copy (overlaps with QK^T WMMA phase)
        for (int it = 0; it < 16; ++it) {
            __builtin_amdgcn_global_load_async_to_lds_b128(
                (AS1 v4i*)(vbase + it * 128 + lane * 4),
                (AS3 v4i*)&vlds[wave][it * 128 + lane * 4], 0, 0);
        }
#endif

        // ---------- scores: (16x16) = Qpad(16x128) * K^T, K-dim chunks of 4
        v8f s8 = (v8f){0,0,0,0,0,0,0,0};
        const float* kptr = kbase + tok * 4 + xoff;           // contiguous 256B/wave
        const float* aptr = qlds + tok * 128 + xoff;          // A: row = lane&15
        for (int dq = 0; dq < 32; ++dq) {
            v2f bv = *(const v2f*)(kptr + dq * 64);
            v2f av = *(const v2f*)(aptr + dq * 4);
            s8 = __builtin_amdgcn_wmma_f32_16x16x4_f32(false, av, false, bv,
                                                       (short)0, s8, false, false);
        }

#if !ASYNC_V
        // ---------- fallback: stage V tile via VGPRs (coalesced b128)
        for (int it = 0; it < 16; ++it) {
            *(v4f*)&vlds[wave][it * 128 + lane * 4] =
                *(const v4f*)(vbase + it * 128 + lane * 4);
        }
#endif

        // ---------- per-head mask + online softmax update
        const int  si = (kb << 4) + tok;
        const bool tv = (si <= past_len);
        unsigned char pb[GQA_] = {p0, p1, p2, p3};
        float pv[GQA_];
        for (int j = 0; j < GQA_; ++j) {
            float s = (tv && pb[j]) ? s8[j] : -INFINITY;
            float r = s;
            for (int sh = 1; sh < 16; sh <<= 1) r = fmaxf(r, __shfl_xor(r, sh, 16));
            float bmax = __shfl(r, 0, 32);
            float mn    = fmaxf(m[j], bmax);
            float alpha = (mn == -INFINITY) ? 1.0f : __expf(m[j] - mn);
            float ph    = (s == -INFINITY) ? 0.0f : __expf(s - mn);
            float rs = ph;
            for (int sh = 1; sh < 16; sh <<= 1) rs += __shfl_xor(rs, sh, 16);
            float rsum = __shfl(rs, 0, 32);
            l[j] = l[j] * alpha + rsum;
            m[j] = mn;
            pv[j] = ph;
            for (int n = 0; n < 8; ++n) acc[n][j] *= alpha;   // rescale head rows
        }

        // ---------- transpose P (head x token) into LDS for WMMA-A layout
        if (lane < 16) {
            for (int j = 0; j < GQA_; ++j) plds[wave][j * 16 + tok] = pv[j];
        }

#if ASYNC_V
        asm volatile("s_wait_asynccnt 0x0" ::: "memory");     // V tile resident in LDS
#endif

        // ---------- acc(16x128) += P(16x16) * V(16x128), token chunks of 4
        const float* pptr = &plds[wave][tok * 16 + xoff];     // A row = lane&15
        for (int tc = 0; tc < 4; ++tc) {
            v2f av = *(const v2f*)(pptr + tc * 4);
            for (int n = 0; n < 8; ++n) {
                v2f bv = *(const v2f*)&vlds[wave][(n * 16 + tok) * 16 + tc * 4 + xoff];
                acc[n] = __builtin_amdgcn_wmma_f32_16x16x4_f32(false, av, false, bv,
                                                               (short)0, acc[n], false, false);
            }
        }
    }

    // ---------- cross-wave softmax merge
    __syncthreads();
    if (lane == 0) {
        for (int j = 0; j < GQA_; ++j) { mx[wave][j] = m[j]; ls[wave][j] = l[j]; }
    }
    __syncthreads();

    float scale[GQA_];
    for (int j = 0; j < GQA_; ++j) {
        float M = -INFINITY;
        for (int w = 0; w < NW_; ++w) M = fmaxf(M, mx[w][j]);
        float L = 0.0f;
        for (int w = 0; w < NW_; ++w)
            L += (mx[w][j] == -INFINITY) ? 0.0f : ls[w][j] * __expf(mx[w][j] - M);
        scale[j] = (m[j] == -INFINITY) ? 0.0f : __expf(m[j] - M) / L;
    }
    if (lane < 16) {
        for (int j = 0; j < GQA_; ++j)
            for (int n = 0; n < 8; ++n)
                atomicAdd(&outbuf[j * 128 + n * 16 + lane], acc[n][j] * scale[j]);
    }
    __syncthreads();

    for (int i = tid; i < GQA_ * 128; i += 32 * NW_) {
        int j = i >> 7, d = i & 127;
        out[((size_t)b * H_ + h0 + j) * D_ + d] = outbuf[i];
    }
}

extern "C" void kernel_launch(void* const* d_in, const int* in_sizes, int n_in,
                              void* d_out, int out_size, void* d_ws, size_t ws_size,
                              hipStream_t stream) {
    const float*         q   = (const float*)d_in[0];
    const float*         k   = (const float*)d_in[1];
    const float*         v   = (const float*)d_in[2];
    const int*           bt  = (const int*)d_in[3];
    const int*           ctx = (const int*)d_in[4];
    const unsigned char* pat = (const unsigned char*)d_in[5];

    dim3 grid(B_ * KV_);
    dim3 block(32 * NW_);
    hipLaunchKernelGGL(sparse_decode_attn, grid, block, 0, stream,
                       q, k, v, bt, ctx, pat, (float*)d_out);
}